// GraphTransformerLayer_52596169507598
// MI455X (gfx1250) — compile-verified
//
#include <hip/hip_runtime.h>

#define ND 64      // channels per head (= model dim)
#define NH 8       // heads
#define HD 512     // NH*ND
#define FF 128     // 2*ND

typedef __attribute__((ext_vector_type(2))) float v2f;
typedef __attribute__((ext_vector_type(8))) float v8f;

// ---------------------------------------------------------------------------
// Float atomic max via ordered integer compare (works for mixed signs).
// ---------------------------------------------------------------------------
__device__ __forceinline__ void atomic_max_f32(float* addr, float val) {
  if (val >= 0.0f) {
    atomicMax((int*)addr, __float_as_int(val));
  } else {
    atomicMin((unsigned int*)addr, __float_as_uint(val));
  }
}

// ---------------------------------------------------------------------------
// Generic fp32 GEMM on the WMMA path: C[M,Ncol] = A[M,K] @ B[K,Ncol] (+bias)(+relu)
// One wave computes one 16x16 tile with V_WMMA_F32_16X16X4_F32, K stepped by 4.
// A 16x4 layout : lanes 0-15 -> (K+0,K+1), lanes 16-31 -> (K+2,K+3), M = lane&15
// B 4x16 layout : lanes 0-15 -> rows K+0,K+1, lanes 16-31 -> rows K+2,K+3, N = lane&15
// C/D 16x16     : VGPR v -> M = v (lanes 0-15) / v+8 (lanes 16-31), N = lane&15
// ---------------------------------------------------------------------------
__global__ void gemm_f32_wmma(const float* __restrict__ A, const float* __restrict__ B,
                              const float* __restrict__ bias, float* __restrict__ C,
                              int K, int Ncol, int relu) {
  const int wave = threadIdx.x >> 5;
  const int lane = threadIdx.x & 31;
  const int half = lane >> 4;
  const int l    = lane & 15;
  const int rowTile   = blockIdx.x;
  const int colTile   = blockIdx.y * 8 + wave;
  const int nColTiles = Ncol >> 4;
  if (colTile >= nColTiles) return;          // wave-uniform: whole wave exits, EXEC stays all-1s

  const int row = rowTile * 16 + l;
  const int col = colTile * 16 + l;
  const float* __restrict__ arow = A + (size_t)row * K;

  v8f acc = {};
  for (int k = 0; k < K; k += 4) {
    v2f a, b;
    a.x = arow[k + 2 * half + 0];
    a.y = arow[k + 2 * half + 1];
    b.x = B[(size_t)(k + 2 * half + 0) * Ncol + col];
    b.y = B[(size_t)(k + 2 * half + 1) * Ncol + col];
    acc = __builtin_amdgcn_wmma_f32_16x16x4_f32(false, a, false, b, (short)0, acc,
                                                false, false);
  }

  const float bcol = bias ? bias[col] : 0.0f;
#pragma unroll
  for (int v = 0; v < 8; ++v) {
    const int r = rowTile * 16 + v + 8 * half;
    float val = acc[v] + bcol;
    if (relu) val = fmaxf(val, 0.0f);
    C[(size_t)r * Ncol + col] = val;
  }
}

// ---------------------------------------------------------------------------
// a_src[n,h] = <h[n,h,:], att_src[h,:]>, a_dst likewise. One thread per (n,h).
// ---------------------------------------------------------------------------
__global__ void attn_scores(const float* __restrict__ h, const float* __restrict__ As,
                            const float* __restrict__ Ad, float* __restrict__ a_src,
                            float* __restrict__ a_dst, int total) {
  int idx = blockIdx.x * blockDim.x + threadIdx.x;
  if (idx >= total) return;
  const int j = idx & (NH - 1);
  const float4* hr = (const float4*)(h + (size_t)idx * ND);
  const float4* ws = (const float4*)(As + j * ND);
  const float4* wd = (const float4*)(Ad + j * ND);
  float s0 = 0.0f, s1 = 0.0f;
#pragma unroll
  for (int c = 0; c < ND / 4; ++c) {
    float4 hv = hr[c], sv = ws[c], dv = wd[c];
    s0 += hv.x * sv.x + hv.y * sv.y + hv.z * sv.z + hv.w * sv.w;
    s1 += hv.x * dv.x + hv.y * dv.y + hv.z * dv.z + hv.w * dv.w;
  }
  a_src[idx] = s0;
  a_dst[idx] = s1;
}

// ---------------------------------------------------------------------------
// Init: m = -inf, s = 0, att_out = 0
// ---------------------------------------------------------------------------
__global__ void init_bufs(float* __restrict__ m, float* __restrict__ s,
                          float* __restrict__ attout, int nm, int na) {
  int i = blockIdx.x * blockDim.x + threadIdx.x;
  if (i < nm) { m[i] = -__builtin_huge_valf(); s[i] = 0.0f; }
  if (i < na) attout[i] = 0.0f;
}

// ---------------------------------------------------------------------------
// Pass A: e = leaky_relu(a_src[src]+a_dst[dst]); store e; atomicMax into m[dst].
// One thread per (edge incl. self-loops).
// ---------------------------------------------------------------------------
__global__ void edge_pass_max(const long long* __restrict__ ei, const float* __restrict__ a_src,
                              const float* __restrict__ a_dst, float* __restrict__ est,
                              float* __restrict__ m, int E, int total) {
  int idx = blockIdx.x * blockDim.x + threadIdx.x;
  if (idx >= total) return;
  int src, dst;
  if (idx < E) { src = (int)ei[idx]; dst = (int)ei[E + idx]; }
  else         { src = idx - E;      dst = idx - E; }
  const float* as = a_src + (size_t)src * NH;
  const float* ad = a_dst + (size_t)dst * NH;
  float* eo = est + (size_t)idx * NH;
#pragma unroll
  for (int j = 0; j < NH; ++j) {
    float e = as[j] + ad[j];
    e = (e > 0.0f) ? e : 0.2f * e;          // leaky_relu(0.2)
    eo[j] = e;
    atomic_max_f32(m + (size_t)dst * NH + j, e);
  }
}

// ---------------------------------------------------------------------------
// Pass B: e = exp(e - m[dst]); store back; atomicAdd into s[dst].
// ---------------------------------------------------------------------------
__global__ void edge_pass_sum(const long long* __restrict__ ei, const float* __restrict__ m,
                              float* __restrict__ est, float* __restrict__ s, int E, int total) {
  int idx = blockIdx.x * blockDim.x + threadIdx.x;
  if (idx >= total) return;
  int dst = (idx < E) ? (int)ei[E + idx] : (idx - E);
  float* eo = est + (size_t)idx * NH;
  const float* mr = m + (size_t)dst * NH;
#pragma unroll
  for (int j = 0; j < NH; ++j) {
    float ex = __expf(eo[j] - mr[j]);
    eo[j] = ex;
    unsafeAtomicAdd(s + (size_t)dst * NH + j, ex);
  }
}

// ---------------------------------------------------------------------------
// Pass C: att_out[dst,c] += (1/H) * sum_h alpha_h * h[src,h,c].
// One wave per edge; lane handles channels (lane, lane+32). h lives in L2 (102MB).
// ---------------------------------------------------------------------------
__global__ void edge_aggregate(const long long* __restrict__ ei, const float* __restrict__ h,
                               const float* __restrict__ est, const float* __restrict__ s,
                               float* __restrict__ attout, int E, int total) {
  const int wave = threadIdx.x >> 5;
  const int lane = threadIdx.x & 31;
  const int idx  = blockIdx.x * 8 + wave;
  if (idx >= total) return;
  int src, dst;
  if (idx < E) { src = (int)ei[idx]; dst = (int)ei[E + idx]; }
  else         { src = idx - E;      dst = idx - E; }
  const float* eo = est + (size_t)idx * NH;
  const float* sr = s + (size_t)dst * NH;
  const float* hb = h + (size_t)src * HD;
  float acc0 = 0.0f, acc1 = 0.0f;
#pragma unroll
  for (int j = 0; j < NH; ++j) {
    const float alpha = eo[j] / (sr[j] + 1e-16f);
    const float* hr = hb + j * ND;
    acc0 += alpha * hr[lane];
    acc1 += alpha * hr[lane + 32];
  }
  unsafeAtomicAdd(attout + (size_t)dst * ND + lane,      acc0 * (1.0f / NH));
  unsafeAtomicAdd(attout + (size_t)dst * ND + lane + 32, acc1 * (1.0f / NH));
}

// ---------------------------------------------------------------------------
// out = LayerNorm(a + b (+ extra_bias)) * g + be.  One wave per row (64 cols),
// lane covers channels (lane, lane+32); butterfly reduce with __shfl_xor.
// ---------------------------------------------------------------------------
__global__ void residual_ln(const float* __restrict__ a, const float* __restrict__ b,
                            const float* __restrict__ extra, const float* __restrict__ g,
                            const float* __restrict__ be, float* __restrict__ out, int Nrows) {
  const int wave = threadIdx.x >> 5;
  const int lane = threadIdx.x & 31;
  const int row  = blockIdx.x * 8 + wave;
  if (row >= Nrows) return;
  const float* ar = a + (size_t)row * ND;
  const float* br = b + (size_t)row * ND;
  float t0 = ar[lane]      + br[lane]      + (extra ? extra[lane]      : 0.0f);
  float t1 = ar[lane + 32] + br[lane + 32] + (extra ? extra[lane + 32] : 0.0f);
  float sum = t0 + t1;
#pragma unroll
  for (int o = 16; o > 0; o >>= 1) sum += __shfl_xor(sum, o, 32);
  const float mu = sum * (1.0f / ND);
  const float d0 = t0 - mu, d1 = t1 - mu;
  float vs = d0 * d0 + d1 * d1;
#pragma unroll
  for (int o = 16; o > 0; o >>= 1) vs += __shfl_xor(vs, o, 32);
  const float inv = rsqrtf(vs * (1.0f / ND) + 1e-5f);
  out[(size_t)row * ND + lane]      = d0 * inv * g[lane]      + be[lane];
  out[(size_t)row * ND + lane + 32] = d1 * inv * g[lane + 32] + be[lane + 32];
}

// ---------------------------------------------------------------------------
extern "C" void kernel_launch(void* const* d_in, const int* in_sizes, int n_in,
                              void* d_out, int out_size, void* d_ws, size_t ws_size,
                              hipStream_t stream) {
  const float*     x        = (const float*)d_in[0];
  const long long* ei       = (const long long*)d_in[1];   // int64 per reference
  const float*     W_att    = (const float*)d_in[2];
  const float*     att_src  = (const float*)d_in[3];
  const float*     att_dst  = (const float*)d_in[4];
  const float*     bias_att = (const float*)d_in[5];
  const float*     W1       = (const float*)d_in[6];
  const float*     b1       = (const float*)d_in[7];
  const float*     W2       = (const float*)d_in[8];
  const float*     b2       = (const float*)d_in[9];
  const float*     ln1_g    = (const float*)d_in[10];
  const float*     ln1_b    = (const float*)d_in[11];
  const float*     ln2_g    = (const float*)d_in[12];
  const float*     ln2_b    = (const float*)d_in[13];
  float*           out      = (float*)d_out;

  const int N  = in_sizes[0] / ND;   // 50000
  const int E  = in_sizes[1] / 2;    // 500000
  const int ET = E + N;              // edges + self loops

  float* ws     = (float*)d_ws;
  float* h      = ws;  ws += (size_t)N * HD;   // [N,H,D]
  float* asrc   = ws;  ws += (size_t)N * NH;
  float* adst   = ws;  ws += (size_t)N * NH;
  float* mbuf   = ws;  ws += (size_t)N * NH;
  float* sbuf   = ws;  ws += (size_t)N * NH;
  float* est    = ws;  ws += (size_t)ET * NH;
  float* attout = ws;  ws += (size_t)N * ND;
  float* x1     = ws;  ws += (size_t)N * ND;
  float* ff1    = ws;  ws += (size_t)N * FF;
  float* ff2    = ws;

  const int rowTiles = N / 16;   // 3125

  // 1) h = x @ W_att                       [N,64] @ [64,512]
  gemm_f32_wmma<<<dim3(rowTiles, (HD / 16) / 8), 256, 0, stream>>>(
      x, W_att, nullptr, h, ND, HD, 0);

  // 2) per-(node,head) attention scores
  attn_scores<<<(N * NH + 255) / 256, 256, 0, stream>>>(h, att_src, att_dst, asrc, adst, N * NH);

  // 3) init m/s/att_out
  init_bufs<<<(N * ND + 255) / 256, 256, 0, stream>>>(mbuf, sbuf, attout, N * NH, N * ND);

  // 4-6) segment softmax + aggregation over E+N edges
  edge_pass_max<<<(ET + 255) / 256, 256, 0, stream>>>(ei, asrc, adst, est, mbuf, E, ET);
  edge_pass_sum<<<(ET + 255) / 256, 256, 0, stream>>>(ei, mbuf, est, sbuf, E, ET);
  edge_aggregate<<<(ET + 7) / 8, 256, 0, stream>>>(ei, h, est, sbuf, attout, E, ET);

  // 7) x1 = LN1(att_out + bias_att + x)
  residual_ln<<<(N + 7) / 8, 256, 0, stream>>>(attout, x, bias_att, ln1_g, ln1_b, x1, N);

  // 8) ff1 = relu(x1 @ W1 + b1)            [N,64] @ [64,128]
  gemm_f32_wmma<<<dim3(rowTiles, 1), 256, 0, stream>>>(x1, W1, b1, ff1, ND, FF, 1);

  // 9) ff2 = ff1 @ W2 + b2                 [N,128] @ [128,64]
  gemm_f32_wmma<<<dim3(rowTiles, 1), 256, 0, stream>>>(ff1, W2, b2, ff2, FF, ND, 0);

  // 10) out = LN2(ff2 + x1)
  residual_ln<<<(N + 7) / 8, 256, 0, stream>>>(ff2, x1, nullptr, ln2_g, ln2_b, out, N);
}